// PointPillarScatter3d_43104291783494
// MI455X (gfx1250) — compile-verified
//
#include <hip/hip_runtime.h>
#include <stdint.h>

// ---------------- problem constants (static per reference) ----------------
#define NXc   360
#define NYc   360
#define Cc    128
#define Bc    4
#define Nc    160000
#define Sc    (NXc * NYc)        // 129600 cells per batch
#define OUTE  (Bc * Cc * Sc)     // 66,355,200 output floats
#define CNTE  (Bc * Sc)          // 518,400 count floats (2 MB in d_ws)

// Tile geometry for the TDM-staged scatter
#define ROWS       32                    // pillar rows per tile
#define TILE_ELEMS (ROWS * Cc)           // 4096 floats = 16 KB per LDS buffer
#define NTILES     (Nc / ROWS)           // 5000 (divides exactly)

typedef __attribute__((ext_vector_type(4))) float         f32x4;
typedef __attribute__((ext_vector_type(4))) unsigned int  u32x4;
typedef __attribute__((ext_vector_type(8))) int           i32x8;
typedef __attribute__((ext_vector_type(4))) int           i32x4;

// ---------------- kernel 1: zero sums (d_out) and counts (d_ws) ----------------
__global__ void ppscat_zero(f32x4* __restrict__ out4, f32x4* __restrict__ cnt4) {
    const int n_out4 = OUTE / 4;              // 16,588,800
    const int n_cnt4 = CNTE / 4;              // 129,600
    const int stride = gridDim.x * blockDim.x;
    f32x4 z = {0.f, 0.f, 0.f, 0.f};
    for (int i = blockIdx.x * blockDim.x + threadIdx.x;
         i < n_out4 + n_cnt4; i += stride) {
        if (i < n_out4) out4[i] = z;
        else            cnt4[i - n_out4] = z;
    }
}

// ---------------- TDM issue: DMA one 32x128 f32 tile into LDS ----------------
// D# built per CDNA5 ISA ch.8: group0 = {flags, lds_addr, global_addr, type=2},
// group1 = {data_size=4B, tensor 128x32, tile 128x32, dim0_stride=128}.
__device__ __forceinline__ void tdm_issue_tile(const float* feat, int tile,
                                               uint32_t lds_off) {
    uint64_t ga = (uint64_t)(uintptr_t)feat +
                  (uint64_t)tile * (uint64_t)(ROWS * Cc * sizeof(float));
    u32x4 g0;
    g0.x = 1u;                                         // count=1 (valid), user mode
    g0.y = lds_off;                                    // LDS byte address
    g0.z = (uint32_t)ga;                               // global_addr[31:0]
    g0.w = (uint32_t)((ga >> 32) & 0x01FFFFFFu)        // global_addr[56:32]
         | (2u << 30);                                 // type = 2 ("image")
    i32x8 g1;
    g1[0] = (int)(2u << 16);                           // data_size = 2 -> 4 bytes
    g1[1] = (int)((uint32_t)Cc   << 16);               // tensor_dim0 = 128 (bits 79:48)
    g1[2] = (int)((uint32_t)ROWS << 16);               // tensor_dim1 = 32  (bits 111:80)
    g1[3] = (int)((uint32_t)Cc   << 16);               // tile_dim0   = 128 (bits 127:112)
    g1[4] = ROWS;                                      // tile_dim1 = 32, tile_dim2 = 0
    g1[5] = Cc;                                        // tensor_dim0_stride = 128
    g1[6] = 0;                                         // stride0 hi / stride1 lo
    g1[7] = 0;                                         // stride1 hi
    i32x4 z4 = {0, 0, 0, 0};                           // groups 2/3 unused (2D tensor)
    i32x8 z8 = {0, 0, 0, 0, 0, 0, 0, 0};               // clang-23 6-arg form extra group
    __builtin_amdgcn_tensor_load_to_lds(g0, g1, z4, z4, z8, 0);
}

// ---------------- kernel 2: double-buffered TDM load + atomic scatter ----------------
__global__ __launch_bounds__(256) void ppscat_scatter(
        const float* __restrict__ feat, const int* __restrict__ coords,
        float* __restrict__ out, float* __restrict__ cnt) {
    __shared__ float buf[2][TILE_ELEMS];   // 2 x 16 KB feature tiles
    __shared__ int   sbase[ROWS];          // per-row transposed output base

    const int tid   = threadIdx.x;
    const bool wave0 = (tid >> 5) == 0;
    const int G     = gridDim.x;

    const uint32_t lds0 = (uint32_t)(uintptr_t)(&buf[0][0]); // LDS offset (low 32b of flat ptr)
    const uint32_t lds1 = (uint32_t)(uintptr_t)(&buf[1][0]);

    int tile = blockIdx.x;
    int cur  = 0;
    if (wave0 && tile < NTILES) tdm_issue_tile(feat, tile, lds0);   // prime pipeline

    for (; tile < NTILES; tile += G) {
        // Per-row cell index + count (independent of feature data -> before the wait)
        if (tid < ROWS) {
            const int row = tile * ROWS + tid;
            const i32x4 c = *(const i32x4*)(coords + 4 * row);      // (b, z, y, x)
            const int b = c[0];
            const int s = c[2] * NXc + c[3];                        // z == 0
            sbase[tid]  = b * (Cc * Sc) + s;                        // out[b, ch, s] base
            atomicAdd(cnt + b * Sc + s, 1.0f);
        }
        const int next = tile + G;
        if (wave0) {
            if (next < NTILES) {
                tdm_issue_tile(feat, next, cur ? lds0 : lds1);      // prefetch next tile
                __builtin_amdgcn_s_wait_tensorcnt(1);               // current tile done
            } else {
                __builtin_amdgcn_s_wait_tensorcnt(0);               // drain
            }
        }
        __syncthreads();                                            // tile data visible

        const float* sb = buf[cur];
        #pragma unroll
        for (int e = tid; e < TILE_ELEMS; e += 256) {
            const int row = e >> 7;          // e / 128
            const int ch  = e & (Cc - 1);    // e % 128
            atomicAdd(out + (size_t)sbase[row] + (size_t)ch * Sc, sb[e]);
        }
        __syncthreads();   // buffer `cur` free before it becomes the next prefetch target
        cur ^= 1;
    }
}

// ---------------- kernel 3: normalize sums -> mean (b128 RMW) ----------------
__global__ void ppscat_norm(f32x4* __restrict__ out4, const float* __restrict__ cnt) {
    const int n4     = OUTE / 4;
    const int rowq   = Cc * Sc / 4;          // float4s per batch image: 4,147,200
    const int sq     = Sc / 4;               // float4s per spatial row: 32,400
    const int stride = gridDim.x * blockDim.x;
    for (int i = blockIdx.x * blockDim.x + threadIdx.x; i < n4; i += stride) {
        const int b  = i / rowq;
        const int s4 = i % sq;                                   // s = 4*s4, 16B aligned
        const f32x4 c4 = *(const f32x4*)(cnt + (size_t)b * Sc + (size_t)s4 * 4);
        f32x4 v = out4[i];
        v.x /= fmaxf(c4.x, 1.0f);
        v.y /= fmaxf(c4.y, 1.0f);
        v.z /= fmaxf(c4.z, 1.0f);
        v.w /= fmaxf(c4.w, 1.0f);
        out4[i] = v;
    }
}

// ---------------- host launch ----------------
extern "C" void kernel_launch(void* const* d_in, const int* in_sizes, int n_in,
                              void* d_out, int out_size, void* d_ws, size_t ws_size,
                              hipStream_t stream) {
    const float* feat   = (const float*)d_in[0];   // [N, 128] f32
    const int*   coords = (const int*)d_in[1];     // [N, 4] i32 (b, z, y, x)
    float* out = (float*)d_out;                    // [B, C, NY, NX] f32
    float* cnt = (float*)d_ws;                     // [B*S] f32 counts (2 MB scratch)

    ppscat_zero   <<<4096, 256, 0, stream>>>((f32x4*)out, (f32x4*)cnt);
    ppscat_scatter<<<1250, 256, 0, stream>>>(feat, coords, out, cnt);
    ppscat_norm   <<<4096, 256, 0, stream>>>((f32x4*)out, cnt);
}